// GraphLoss_52544629899414
// MI455X (gfx1250) — compile-verified
//
#include <hip/hip_runtime.h>
#include <math.h>

// ---------------------------------------------------------------------------
// GraphLoss on MI455X (gfx1250).
//
// Reference reduces to:
//   e1[j]   = -w[j]                                   (j = 0..63)
//   e_{l+1} = LSE_i( e_l[i] - wB[l][i][j] )           (l = 0..62, wB row-major 64x64)
//   final   = LSE_i( e_64[i] - w[E-64+i] )
//   out     = dot(weight, gold) + final               (single f32 scalar)
//
// gold dot: bandwidth bound (~4MB @ 23.3TB/s) -> multi-block partial sums.
// lattice:  latency bound, 63 sequential 64x64 semiring matvecs. With a
// per-level max rescale it becomes a plain f32 matvec:
//   e'[j] = M + log( sum_i exp(e_i - M) * exp(-wB[i][j]) )
// run through V_WMMA_F32_16X16X4_F32 on a single wave32, with each level's
// 16KB weight tile double-buffered into LDS via GLOBAL_LOAD_ASYNC_TO_LDS_B128
// (ASYNCcnt completes in order, so asynccnt<=32 == "current tile ready" while
// the next tile is still streaming).
// ---------------------------------------------------------------------------

typedef __attribute__((ext_vector_type(2))) float v2f;
typedef __attribute__((ext_vector_type(8))) float v8f;

#define GOLD_BLOCKS  256
#define GOLD_THREADS 256

// Kernel 1: deterministic per-block partial sums of weight[e]*gold[e].
__global__ __launch_bounds__(GOLD_THREADS)
void gold_dot_kernel(const float* __restrict__ w, const int* __restrict__ graph,
                     float* __restrict__ partials, int E) {
    __shared__ float red[GOLD_THREADS];
    const int tid    = blockIdx.x * blockDim.x + threadIdx.x;
    const int stride = gridDim.x * blockDim.x;
    float s = 0.0f;
    for (int e = tid; e < E; e += stride) {
        s += w[e] * (float)graph[3 * e + 2];   // graph is int32 [E,3]; col 2 = gold
    }
    red[threadIdx.x] = s;
    __syncthreads();
    for (int off = GOLD_THREADS / 2; off > 0; off >>= 1) {
        if (threadIdx.x < off) red[threadIdx.x] += red[threadIdx.x + off];
        __syncthreads();
    }
    if (threadIdx.x == 0) partials[blockIdx.x] = red[0];
}

// Issue one level tile (64x64 f32 = 16KB) as 32 wave-wide async B128 copies.
// Each instruction: 32 lanes x 16B = 512B. Generic LDS pointers carry the
// wave-relative LDS byte offset in their low 32 bits (aperture layout), which
// is exactly what the VDST operand of global_load_async_to_lds wants.
__device__ __forceinline__ void async_copy_tile(float* ldsDst, const float* gSrc,
                                                int lane) {
#pragma unroll
    for (int c = 0; c < 32; ++c) {
        const unsigned ldsAddr =
            (unsigned)(uintptr_t)(ldsDst + c * 128 + lane * 4);
        const float* g = gSrc + c * 128 + lane * 4;
        asm volatile("global_load_async_to_lds_b128 %0, %1, off"
                     :: "v"(ldsAddr), "v"(g)
                     : "memory");
    }
}

__device__ __forceinline__ void wait_async_le32() {
    asm volatile("s_wait_asynccnt 0x20" ::: "memory");
}
__device__ __forceinline__ void wait_async_0() {
    asm volatile("s_wait_asynccnt 0x0" ::: "memory");
}

// Kernel 2: single-wave WMMA lattice recurrence + final reduction.
//
// WMMA f32 16x16x4 fragment layouts (ISA 7.12.2, wave32):
//   A 16x4 : VGPR0 = {K=0 lanes 0-15 | K=2 lanes 16-31}, VGPR1 = {K=1 | K=3}
//            (rows M = lane % 16; we broadcast the 'a' vector into all rows)
//   B 4x16 : mirrored, N = lane % 16 per half, VGPR0 = {K=0 | K=2}, VGPR1 = {K=1 | K=3}
//   D 16x16: VGPR0 lanes 0-15 = row M=0 (our matvec result), N = lane
__global__ __launch_bounds__(32)
void lattice_wmma_kernel(const float* __restrict__ w, int E,
                         const float* __restrict__ partials, int nPartials,
                         float* __restrict__ out) {
    __shared__ float eL[64];            // log-domain layer state
    __shared__ float aL[64];            // rescaled exp-domain state
    __shared__ float tile[2][4096];     // double-buffered 64x64 level weights

    const int lane  = threadIdx.x;      // 0..31, one wave
    const int off   = (lane >> 4) << 1; // 0 for lanes 0-15, 2 for lanes 16-31
    const int col16 = lane & 15;

    // Level 1: e[j] = -w[j]
    eL[lane]      = -w[lane];
    eL[lane + 32] = -w[lane + 32];

    const float* wB = w + 64;           // 63 bipartite 64x64 blocks, row-major

    // Prime the pipeline: stream level-0 tile into buffer 0.
    async_copy_tile(&tile[0][0], wB, lane);

    for (int l = 0; l < 63; ++l) {
        float* cur = &tile[l & 1][0];
        const bool more = (l + 1 < 63);
        if (more) async_copy_tile(&tile[(l + 1) & 1][0], wB + (size_t)(l + 1) * 4096, lane);

        // ---- rescale: M = max(e), a = exp(e - M) ----
        const float e0 = eL[lane];
        const float e1 = eL[lane + 32];
        float m = fmaxf(e0, e1);
        for (int d = 16; d > 0; d >>= 1) m = fmaxf(m, __shfl_xor(m, d, 32));
        aL[lane]      = __expf(e0 - m);
        aL[lane + 32] = __expf(e1 - m);
        __syncthreads();

        // Current tile guaranteed resident: async completes in order, so with
        // the next tile's 32 copies in flight, <=32 outstanding == ours done.
        if (more) wait_async_le32(); else wait_async_0();

        // ---- A fragments for all 16 K-chunks (row-broadcast of 'a') ----
        float A0[16], A1[16];
#pragma unroll
        for (int k = 0; k < 16; ++k) {
            A0[k] = aL[4 * k + off];
            A1[k] = aL[4 * k + off + 1];
        }

        // ---- t(1x64) = a(1x64) x exp(-tile)(64x64) via 4x16 WMMA ops ----
#pragma unroll
        for (int n = 0; n < 4; ++n) {
            v8f acc = {0.f, 0.f, 0.f, 0.f, 0.f, 0.f, 0.f, 0.f};
            const int colg = n * 16 + col16;
#pragma unroll
            for (int k = 0; k < 16; ++k) {
                const int r = 4 * k + off;
                v2f A, B;
                A.x = A0[k];
                A.y = A1[k];
                B.x = __expf(-cur[r * 64 + colg]);
                B.y = __expf(-cur[(r + 1) * 64 + colg]);
                // D = A x B + C, full f32 (8 args: neg_a,A,neg_b,B,c_mod,C,reuse_a,reuse_b)
                acc = __builtin_amdgcn_wmma_f32_16x16x4_f32(
                    false, A, false, B, (short)0, acc, false, false);
            }
            // Row 0 of D lives in acc[0] on lanes 0-15 (N = lane).
            if (lane < 16) eL[colg] = m + __logf(acc[0]);
        }
        __syncthreads();
    }

    // ---- sink: final = LSE_i( e[i] - w_sink[i] ) ----
    const float* wsink = w + (E - 64);
    const float v0 = eL[lane]      - wsink[lane];
    const float v1 = eL[lane + 32] - wsink[lane + 32];
    float m = fmaxf(v0, v1);
    for (int d = 16; d > 0; d >>= 1) m = fmaxf(m, __shfl_xor(m, d, 32));
    float s = __expf(v0 - m) + __expf(v1 - m);
    for (int d = 16; d > 0; d >>= 1) s += __shfl_xor(s, d, 32);

    // ---- gold score: finish the partial sums ----
    float g = 0.0f;
    for (int i = lane; i < nPartials; i += 32) g += partials[i];
    for (int d = 16; d > 0; d >>= 1) g += __shfl_xor(g, d, 32);

    if (lane == 0) out[0] = g + m + __logf(s);
}

extern "C" void kernel_launch(void* const* d_in, const int* in_sizes, int n_in,
                              void* d_out, int out_size, void* d_ws, size_t ws_size,
                              hipStream_t stream) {
    const int*   graph = (const int*)d_in[0];     // int32 [E,3] (src, dst, gold)
    const float* w     = (const float*)d_in[1];   // float32 [E]
    float*       out   = (float*)d_out;           // float32 [1]
    float*       parts = (float*)d_ws;            // >= GOLD_BLOCKS floats of scratch

    const int E = in_sizes[1];

    int blocks = GOLD_BLOCKS;
    const int maxp = (int)(ws_size / sizeof(float));
    if (maxp > 0 && maxp < blocks) blocks = maxp;
    if (blocks < 1) blocks = 1;

    gold_dot_kernel<<<blocks, GOLD_THREADS, 0, stream>>>(w, graph, parts, E);
    lattice_wmma_kernel<<<1, 32, 0, stream>>>(w, E, parts, blocks, out);
}